// mamba_1024_17360257810979
// MI455X (gfx1250) — compile-verified
//
#include <hip/hip_runtime.h>
#include <hip/hip_bf16.h>
#include <cstdint>

// ---------------------------------------------------------------------------
// Vision-Mamba hybrid block forward for gfx1250 (MI455X).
// All GEMMs routed through V_WMMA_F32_16X16X4_F32 (fp32 in/out, matches the
// fp32 reference exactly in precision class). 32x32 tile per wave32.
// ---------------------------------------------------------------------------

typedef float v2f __attribute__((ext_vector_type(2)));
typedef float v8f __attribute__((ext_vector_type(8)));

#define D_MODEL 512
#define D_INNER 256
#define D_STATE 16
#define DT_RANK 16
#define NH      8
#define HD      64
#define BATCH   32
#define LTOK    196
#define MROWS   (BATCH * LTOK)   // 6272

__device__ __forceinline__ v8f wmma4(v2f a, v2f b, v8f c) {
  // D = A(16x4, f32) * B(4x16, f32) + C(16x16, f32)
  return __builtin_amdgcn_wmma_f32_16x16x4_f32(
      /*neg_a=*/false, a, /*neg_b=*/false, b,
      /*c_mod=*/(short)0, c, /*reuse_a=*/false, /*reuse_b=*/false);
}

__device__ __forceinline__ float act_gelu(float x) {
  // jax.nn.gelu default (tanh approximation)
  float x3 = x * x * x;
  return 0.5f * x * (1.0f + tanhf(0.7978845608028654f * (x + 0.044715f * x3)));
}
__device__ __forceinline__ float act_softplus(float x) {
  return (x > 20.0f) ? x : log1pf(__expf(x));
}
__device__ __forceinline__ float act_silu(float x) {
  return x / (1.0f + __expf(-x));
}

// ---------------------------------------------------------------------------
// Generic fp32 WMMA GEMM:  C[M,N] = act(A[M,lda] @ W[N,K]^T + bias) (+ res)
// One wave computes a 32x32 tile of C via 4 accumulators (2 m-frags x 2 n-frags).
// ACT: 0=none, 1=gelu, 2=softplus
// ---------------------------------------------------------------------------
template <int ACT, bool HAS_BIAS, bool HAS_RES>
__global__ void gemm_f32_wmma(const float* __restrict__ A, int lda,
                              const float* __restrict__ W,
                              const float* __restrict__ bias,
                              const float* __restrict__ res,
                              float* __restrict__ C,
                              int M, int N, int K) {
  const int lane   = threadIdx.x & 31;
  const int wave   = threadIdx.x >> 5;
  const int tilesN = (N + 31) >> 5;
  const int tilesM = (M + 31) >> 5;
  const int gw     = blockIdx.x * (blockDim.x >> 5) + wave;
  const int tm     = gw / tilesN;
  const int tn     = gw - tm * tilesN;
  if (tm >= tilesM) return;                    // wave-uniform exit

  const int half = lane >> 4;                  // 0..1
  const int idx  = lane & 15;                  // 0..15

  const int r0 = tm * 32 + idx;                // A rows for frag 0
  const int r1 = r0 + 16;                      // A rows for frag 1
  const int c0 = tn * 32 + idx;                // B cols (weight rows) frag 0
  const int c1 = c0 + 16;

  const bool vr0 = r0 < M, vr1 = r1 < M, vc0 = c0 < N, vc1 = c1 < N;
  const float* a0p = A + (size_t)(vr0 ? r0 : 0) * lda;
  const float* a1p = A + (size_t)(vr1 ? r1 : 0) * lda;
  const float* b0p = W + (size_t)(vc0 ? c0 : 0) * K;
  const float* b1p = W + (size_t)(vc1 ? c1 : 0) * K;

  const v2f z2 = {0.0f, 0.0f};
  v8f acc00 = {0.f,0.f,0.f,0.f,0.f,0.f,0.f,0.f};
  v8f acc01 = acc00, acc10 = acc00, acc11 = acc00;

  for (int k = 0; k < K; k += 4) {
    const int kk = k + half * 2;               // even -> 8B aligned float2
    v2f a0 = vr0 ? *(const v2f*)(a0p + kk) : z2;
    v2f a1 = vr1 ? *(const v2f*)(a1p + kk) : z2;
    v2f b0 = vc0 ? *(const v2f*)(b0p + kk) : z2;
    v2f b1 = vc1 ? *(const v2f*)(b1p + kk) : z2;
    acc00 = wmma4(a0, b0, acc00);
    acc01 = wmma4(a0, b1, acc01);
    acc10 = wmma4(a1, b0, acc10);
    acc11 = wmma4(a1, b1, acc11);
  }

  // Store: C/D frag layout -> row = mbase + r + 8*half, col = nbase + idx.
#define STORE_TILE(ACCV, MBASE, COL, VC)                                   \
  if (VC) {                                                                \
    const float bv = HAS_BIAS ? bias[COL] : 0.0f;                          \
    _Pragma("unroll")                                                      \
    for (int r = 0; r < 8; ++r) {                                          \
      const int row = (MBASE) + half * 8 + r;                              \
      if (row < M) {                                                       \
        float v = (ACCV)[r] + bv;                                          \
        if (ACT == 1) v = act_gelu(v);                                     \
        else if (ACT == 2) v = act_softplus(v);                            \
        if (HAS_RES) v += res[(size_t)row * N + (COL)];                    \
        C[(size_t)row * N + (COL)] = v;                                    \
      }                                                                    \
    }                                                                      \
  }
  STORE_TILE(acc00, tm * 32,      c0, vc0)
  STORE_TILE(acc01, tm * 32,      c1, vc1)
  STORE_TILE(acc10, tm * 32 + 16, c0, vc0)
  STORE_TILE(acc11, tm * 32 + 16, c1, vc1)
#undef STORE_TILE
}

// ---------------------------------------------------------------------------
// Patch embedding as an implicit im2col WMMA GEMM.
// t[m, o] = sum_{k<768} x_im2col[m, k] * patch_w[o, k] + patch_b[o]
//   m = b*196 + l, l = py*14 + px ; k = c*256 + ky*16 + kx
// M = 6272 (multiple of 32), N = 512, K = 768 -> no bounds checks needed.
// ---------------------------------------------------------------------------
__global__ void patch_embed_wmma(const float* __restrict__ x,
                                 const float* __restrict__ W,
                                 const float* __restrict__ bias,
                                 float* __restrict__ C) {
  constexpr int M = MROWS, N = D_MODEL, K = 768;
  const int lane   = threadIdx.x & 31;
  const int wave   = threadIdx.x >> 5;
  const int tilesN = N / 32;                   // 16
  const int gw     = blockIdx.x * (blockDim.x >> 5) + wave;
  const int tm     = gw / tilesN;
  const int tn     = gw - tm * tilesN;
  if (tm >= M / 32) return;

  const int half = lane >> 4;
  const int idx  = lane & 15;

  // Decode the two A rows this lane feeds.
  const int r0 = tm * 32 + idx;
  const int r1 = r0 + 16;
  const int b0i = r0 / LTOK, l0 = r0 - b0i * LTOK;
  const int b1i = r1 / LTOK, l1 = r1 - b1i * LTOK;
  const int py0 = l0 / 14, px0 = l0 - py0 * 14;
  const int py1 = l1 / 14, px1 = l1 - py1 * 14;
  const int base0c = b0i * 3, base1c = b1i * 3;
  const int y0 = py0 * 16, x0c = px0 * 16;
  const int y1 = py1 * 16, x1c = px1 * 16;

  const int c0 = tn * 32 + idx;
  const int c1 = c0 + 16;
  const float* b0p = W + (size_t)c0 * K;
  const float* b1p = W + (size_t)c1 * K;

  v8f acc00 = {0.f,0.f,0.f,0.f,0.f,0.f,0.f,0.f};
  v8f acc01 = acc00, acc10 = acc00, acc11 = acc00;

  for (int k = 0; k < K; k += 4) {
    const int kk  = k + half * 2;              // even
    const int c   = kk >> 8;                   // channel
    const int rem = kk & 255;
    const int ky  = rem >> 4;
    const int kx  = rem & 15;                  // even -> (kx,kx+1) contiguous
    v2f a0 = *(const v2f*)(x + (((size_t)(base0c + c) * 224 + y0 + ky) * 224 + x0c + kx));
    v2f a1 = *(const v2f*)(x + (((size_t)(base1c + c) * 224 + y1 + ky) * 224 + x1c + kx));
    v2f b0 = *(const v2f*)(b0p + kk);
    v2f b1 = *(const v2f*)(b1p + kk);
    acc00 = wmma4(a0, b0, acc00);
    acc01 = wmma4(a0, b1, acc01);
    acc10 = wmma4(a1, b0, acc10);
    acc11 = wmma4(a1, b1, acc11);
  }

  const float bv0 = bias[c0], bv1 = bias[c1];
#pragma unroll
  for (int r = 0; r < 8; ++r) {
    const int row0 = tm * 32 + half * 8 + r;
    const int row1 = row0 + 16;
    C[(size_t)row0 * N + c0] = acc00[r] + bv0;
    C[(size_t)row0 * N + c1] = acc01[r] + bv1;
    C[(size_t)row1 * N + c0] = acc10[r] + bv0;
    C[(size_t)row1 * N + c1] = acc11[r] + bv1;
  }
}

// ---------------------------------------------------------------------------
// LayerNorm over rows of 512 (one block of 256 threads per row).
// ---------------------------------------------------------------------------
__global__ void layernorm512(const float* __restrict__ in,
                             const float* __restrict__ g,
                             const float* __restrict__ b,
                             float* __restrict__ out) {
  __shared__ float ssum[256];
  __shared__ float ssq[256];
  const int row = blockIdx.x;
  const int tid = threadIdx.x;
  const float x0 = in[(size_t)row * 512 + tid];
  const float x1 = in[(size_t)row * 512 + tid + 256];
  ssum[tid] = x0 + x1;
  ssq[tid]  = x0 * x0 + x1 * x1;
  __syncthreads();
  for (int s = 128; s > 0; s >>= 1) {
    if (tid < s) { ssum[tid] += ssum[tid + s]; ssq[tid] += ssq[tid + s]; }
    __syncthreads();
  }
  const float mean = ssum[0] * (1.0f / 512.0f);
  const float var  = ssq[0] * (1.0f / 512.0f) - mean * mean;
  const float inv  = rsqrtf(var + 1e-6f);
  out[(size_t)row * 512 + tid]       = (x0 - mean) * inv * g[tid]       + b[tid];
  out[(size_t)row * 512 + tid + 256] = (x1 - mean) * inv * g[tid + 256] + b[tid + 256];
}

// ---------------------------------------------------------------------------
// Causal depthwise conv (k=4, left pad 3) + SiLU for both xm and z halves.
// xz: [M, 512] (cols 0..255 = xm, 256..511 = z)
// xmc: [M, 256] = silu(conv(xm)) ; yz[:, 256:512] = silu(conv(z))
// ---------------------------------------------------------------------------
__global__ void dwconv_silu(const float* __restrict__ xz,
                            const float* __restrict__ cxw,
                            const float* __restrict__ cxb,
                            const float* __restrict__ czw,
                            const float* __restrict__ czb,
                            float* __restrict__ xmc,
                            float* __restrict__ yz) {
  const int id = blockIdx.x * blockDim.x + threadIdx.x;
  if (id >= BATCH * LTOK * D_INNER) return;
  const int d = id & (D_INNER - 1);
  const int l = (id >> 8) % LTOK;
  const int b = id / (LTOK * D_INNER);
  const int m = b * LTOK + l;
  float sx = cxb[d], sz = czb[d];
#pragma unroll
  for (int j = 0; j < 4; ++j) {
    const int ls = l - 3 + j;
    if (ls >= 0) {
      const size_t src = (size_t)(b * LTOK + ls) * 512;
      sx += cxw[d * 4 + j] * xz[src + d];
      sz += czw[d * 4 + j] * xz[src + 256 + d];
    }
  }
  xmc[(size_t)m * 256 + d]      = act_silu(sx);
  yz[(size_t)m * 512 + 256 + d] = act_silu(sz);
}

// ---------------------------------------------------------------------------
// Selective scan: one thread per (b, d); 16 states in registers, 196 steps.
// yz[:, 0:256] = y + xm * Dp
// ---------------------------------------------------------------------------
__global__ void ssm_scan(const float* __restrict__ dtb,
                         const float* __restrict__ xdbl,
                         const float* __restrict__ xmc,
                         const float* __restrict__ A_log,
                         const float* __restrict__ Dp,
                         float* __restrict__ yz) {
  const int id = blockIdx.x * blockDim.x + threadIdx.x;
  if (id >= BATCH * D_INNER) return;
  const int d = id & (D_INNER - 1);
  const int b = id >> 8;
  float Areg[D_STATE], h[D_STATE];
#pragma unroll
  for (int n = 0; n < D_STATE; ++n) {
    Areg[n] = -__expf(A_log[d * D_STATE + n]);
    h[n] = 0.0f;
  }
  const float dval = Dp[d];
  for (int t = 0; t < LTOK; ++t) {
    const int m = b * LTOK + t;
    const float dtv = dtb[(size_t)m * D_INNER + d];
    const float u   = xmc[(size_t)m * D_INNER + d];
    const float du  = dtv * u;
    const float* Brow = xdbl + (size_t)m * 48 + DT_RANK;
    const float* Crow = Brow + D_STATE;
    float y = 0.0f;
#pragma unroll
    for (int n = 0; n < D_STATE; ++n) {
      const float dA = __expf(dtv * Areg[n]);
      h[n] = dA * h[n] + du * Brow[n];
      y += h[n] * Crow[n];
    }
    yz[(size_t)m * 512 + d] = y + u * dval;
  }
}

// ---------------------------------------------------------------------------
// Attention: online-softmax, one q-row per thread. qkv: [M, 1536] as
// [q(8x64) | k(8x64) | v(8x64)] per token.  grid = B*NH*ceil(196/64).
// ---------------------------------------------------------------------------
__global__ void attention_fwd(const float* __restrict__ qkv,
                              float* __restrict__ ob) {
  const int bh    = blockIdx.x >> 2;
  const int chunk = blockIdx.x & 3;
  const int b     = bh >> 3;
  const int h     = bh & 7;
  const int qi    = chunk * 64 + threadIdx.x;
  const bool active = qi < LTOK;

  float q[HD], acc[HD];
  const float* qp = qkv + (size_t)(b * LTOK + (active ? qi : 0)) * 1536 + h * HD;
#pragma unroll
  for (int e = 0; e < HD; ++e) { q[e] = qp[e] * 0.125f; acc[e] = 0.0f; }

  float mmax = -1e30f, ssum = 0.0f;
  for (int kk = 0; kk < LTOK; ++kk) {
    const float* kp = qkv + (size_t)(b * LTOK + kk) * 1536 + 512 + h * HD;
    const float* vp = kp + 512;
    float s = 0.0f;
#pragma unroll
    for (int e = 0; e < HD; ++e) s += q[e] * kp[e];
    const float mnew = fmaxf(mmax, s);
    const float corr = __expf(mmax - mnew);
    const float p    = __expf(s - mnew);
    ssum = ssum * corr + p;
#pragma unroll
    for (int e = 0; e < HD; ++e) acc[e] = acc[e] * corr + p * vp[e];
    mmax = mnew;
  }
  if (active) {
    const float inv = 1.0f / ssum;
    float* op = ob + (size_t)(b * LTOK + qi) * 512 + h * HD;
#pragma unroll
    for (int e = 0; e < HD; ++e) op[e] = acc[e] * inv;
  }
}

// ---------------------------------------------------------------------------
// Mean pool over L (the head linear commutes with the mean).
// ---------------------------------------------------------------------------
__global__ void meanpool512(const float* __restrict__ t, float* __restrict__ p) {
  const int id = blockIdx.x * blockDim.x + threadIdx.x;
  if (id >= BATCH * 512) return;
  const int d = id & 511;
  const int b = id >> 9;
  float s = 0.0f;
  for (int l = 0; l < LTOK; ++l) s += t[(size_t)(b * LTOK + l) * 512 + d];
  p[(size_t)b * 512 + d] = s * (1.0f / (float)LTOK);
}

// Final tiny fc: out[b, j] = pooled_feat[b,:] . fc_w[j,:] + fc_b[j]
__global__ void fc_small(const float* __restrict__ pf,
                         const float* __restrict__ w,
                         const float* __restrict__ bias,
                         float* __restrict__ out) {
  const int id = blockIdx.x * blockDim.x + threadIdx.x;
  if (id >= BATCH * 4) return;
  const int j = id & 3;
  const int b = id >> 2;
  float s = bias[j];
  for (int k = 0; k < 1024; ++k) s += pf[(size_t)b * 1024 + k] * w[(size_t)j * 1024 + k];
  out[(size_t)b * 4 + j] = s;
}

// ---------------------------------------------------------------------------
// Launcher
// ---------------------------------------------------------------------------
static inline int gemm_grid(int M, int N) {
  const int waves = ((M + 31) / 32) * ((N + 31) / 32);
  return (waves + 3) / 4;  // 4 waves (128 threads) per block
}

extern "C" void kernel_launch(void* const* d_in, const int* in_sizes, int n_in,
                              void* d_out, int out_size, void* d_ws, size_t ws_size,
                              hipStream_t stream) {
  const float* x           = (const float*)d_in[0];
  const float* patch_w     = (const float*)d_in[1];
  const float* patch_b     = (const float*)d_in[2];
  const float* ln1_g       = (const float*)d_in[3];
  const float* ln1_b       = (const float*)d_in[4];
  const float* in_proj_w   = (const float*)d_in[5];
  const float* convx_w     = (const float*)d_in[6];
  const float* convx_b     = (const float*)d_in[7];
  const float* convz_w     = (const float*)d_in[8];
  const float* convz_b     = (const float*)d_in[9];
  const float* x_proj_w    = (const float*)d_in[10];
  const float* dt_proj_w   = (const float*)d_in[11];
  const float* dt_proj_b   = (const float*)d_in[12];
  const float* A_log       = (const float*)d_in[13];
  const float* Dp          = (const float*)d_in[14];
  const float* out_proj_w  = (const float*)d_in[15];
  const float* ln2_g       = (const float*)d_in[16];
  const float* ln2_b       = (const float*)d_in[17];
  const float* qkv_w       = (const float*)d_in[18];
  const float* attn_proj_w = (const float*)d_in[19];
  const float* ln3_g       = (const float*)d_in[20];
  const float* ln3_b       = (const float*)d_in[21];
  const float* mlp_w1      = (const float*)d_in[22];
  const float* mlp_b1      = (const float*)d_in[23];
  const float* mlp_w2      = (const float*)d_in[24];
  const float* mlp_b2      = (const float*)d_in[25];
  const float* head_w      = (const float*)d_in[26];
  const float* head_b      = (const float*)d_in[27];
  const float* fc_w        = (const float*)d_in[28];
  const float* fc_b        = (const float*)d_in[29];
  float* out = (float*)d_out;

  // Workspace layout (bytes); phases reuse the big unions.  Peak ~104 MB.
  char* ws = (char*)d_ws;
  const size_t SZ_T = (size_t)MROWS * 512 * 4;      // 12,845,056
  float* t    = (float*)(ws + 0);
  float* hbuf = (float*)(ws + SZ_T);
  char*  U1   = ws + 2 * SZ_T;                      // max(xz,qkv,mid) = 51.4 MB
  char*  U2   = U1 + (size_t)MROWS * 2048 * 4;
  float* xz   = (float*)U1;                         // [M,512]
  float* qkvb = (float*)U1;                         // [M,1536] (after xz dead)
  float* mid  = (float*)U1;                         // [M,2048] (after qkv dead)
  float* featp= (float*)U1;                         // [32,1024] (after mid dead)
  float* xmc  = (float*)(U2 + 0);                   // [M,256]
  float* xdbl = (float*)(U2 + (size_t)MROWS * 256 * 4);            // [M,48]
  float* dtb  = (float*)(U2 + (size_t)MROWS * 256 * 4
                            + (size_t)MROWS * 48 * 4);             // [M,256]
  float* yzb  = (float*)(U2 + (size_t)MROWS * 560 * 4);            // [M,512]
  float* ob   = (float*)(U2 + 0);                   // [M,512] (mamba bufs dead)
  float* ptl  = (float*)(U2 + 0);                   // [32,512] pooled tokens

  const int M = MROWS;

  // 1) Patch embed -> t  (WMMA GEMM, M=6272 N=512 K=768)
  patch_embed_wmma<<<gemm_grid(M, 512), 128, 0, stream>>>(x, patch_w, patch_b, t);
  // 2) LN1 -> h
  layernorm512<<<M, 256, 0, stream>>>(t, ln1_g, ln1_b, hbuf);
  // 3) in_proj: xz = h @ in_proj_w^T   (N=512, K=512)
  gemm_f32_wmma<0, false, false><<<gemm_grid(M, 512), 128, 0, stream>>>(
      hbuf, 512, in_proj_w, nullptr, nullptr, xz, M, 512, 512);
  // 4) causal depthwise convs + SiLU -> xmc, yz[:,256:]
  dwconv_silu<<<(M * 256 + 255) / 256, 256, 0, stream>>>(
      xz, convx_w, convx_b, convz_w, convz_b, xmc, yzb);
  // 5) x_proj: xdbl = xmc @ x_proj_w^T  (N=48, K=256)
  gemm_f32_wmma<0, false, false><<<gemm_grid(M, 48), 128, 0, stream>>>(
      xmc, 256, x_proj_w, nullptr, nullptr, xdbl, M, 48, 256);
  // 6) dt = softplus(xdbl[:, :16] @ dt_proj_w^T + dt_proj_b)  (N=256, K=16, lda=48)
  gemm_f32_wmma<2, true, false><<<gemm_grid(M, 256), 128, 0, stream>>>(
      xdbl, 48, dt_proj_w, dt_proj_b, nullptr, dtb, M, 256, 16);
  // 7) selective scan -> yz[:, :256]
  ssm_scan<<<(BATCH * D_INNER + 255) / 256, 256, 0, stream>>>(
      dtb, xdbl, xmc, A_log, Dp, yzb);
  // 8) out_proj + residual: t += yz @ out_proj_w^T  (N=512, K=512)
  gemm_f32_wmma<0, false, true><<<gemm_grid(M, 512), 128, 0, stream>>>(
      yzb, 512, out_proj_w, nullptr, t, t, M, 512, 512);
  // 9) LN2 -> h ; qkv = h @ qkv_w^T  (N=1536, K=512)
  layernorm512<<<M, 256, 0, stream>>>(t, ln2_g, ln2_b, hbuf);
  gemm_f32_wmma<0, false, false><<<gemm_grid(M, 1536), 128, 0, stream>>>(
      hbuf, 512, qkv_w, nullptr, nullptr, qkvb, M, 1536, 512);
  // 10) attention -> ob
  attention_fwd<<<BATCH * NH * 4, 64, 0, stream>>>(qkvb, ob);
  // 11) attn_proj + residual: t += ob @ attn_proj_w^T
  gemm_f32_wmma<0, false, true><<<gemm_grid(M, 512), 128, 0, stream>>>(
      ob, 512, attn_proj_w, nullptr, t, t, M, 512, 512);
  // 12) LN3 -> h ; MLP
  layernorm512<<<M, 256, 0, stream>>>(t, ln3_g, ln3_b, hbuf);
  gemm_f32_wmma<1, true, false><<<gemm_grid(M, 2048), 128, 0, stream>>>(
      hbuf, 512, mlp_w1, mlp_b1, nullptr, mid, M, 2048, 512);
  gemm_f32_wmma<0, true, true><<<gemm_grid(M, 512), 128, 0, stream>>>(
      mid, 2048, mlp_w2, mlp_b2, t, t, M, 512, 2048);
  // 13) mean-pool tokens, then head linear on pooled (mean commutes with linear)
  meanpool512<<<(BATCH * 512 + 255) / 256, 256, 0, stream>>>(t, ptl);
  gemm_f32_wmma<0, true, false><<<gemm_grid(32, 1024), 128, 0, stream>>>(
      ptl, 512, head_w, head_b, nullptr, featp, 32, 1024, 512);
  // 14) fc -> out [32,4]
  fc_small<<<1, 128, 0, stream>>>(featp, fc_w, fc_b, out);
}